// SpatioTemporalAttention_51470888075371
// MI455X (gfx1250) — compile-verified
//
#include <hip/hip_runtime.h>
#include <math.h>

// ---------------- problem dims ----------------
#define BB   16
#define TT   24
#define NN   128
#define DD   64
#define KIN  192                  // 3*D input channels of XE
#define PTOT (BB*TT*NN)           // 49152 positions
#define SCALE 0.35355339059327373f // 1/sqrt(8)

typedef __attribute__((ext_vector_type(2))) float v2f;
typedef __attribute__((ext_vector_type(8))) float v8f;

// D = A(16x4) * B(4x16) + C, fp32. 8-arg VOP3P form:
// (neg_a, A, neg_b, B, c_mod, C, reuse_a, reuse_b)
__device__ __forceinline__ v8f wmma4(v2f a, v2f b, v8f c) {
  return __builtin_amdgcn_wmma_f32_16x16x4_f32(false, a, false, b, (short)0, c,
                                               false, false);
}

// Async global -> LDS copy (16B per lane), tracked by ASYNCcnt.
// LDS address = low 32 bits of the generic shared pointer (WG-relative).
__device__ __forceinline__ void async_b128(void* lds, const void* g) {
  unsigned l = (unsigned)(uintptr_t)lds;
  asm volatile("global_load_async_to_lds_b128 %0, %1, off"
               :
               : "v"(l), "v"(g)
               : "memory");
}
__device__ __forceinline__ void async_wait0() {
  asm volatile("s_wait_asynccnt 0" ::: "memory");
}

// =====================================================================
// Kernel 1: six fused projections  relu(XE @ W + b),  XE = [X | TLE]
// grid = PTOT/16 = 3072 blocks, block = 192 threads (6 waves, 1 per proj)
// =====================================================================
__global__ __launch_bounds__(192) void proj_kernel(
    const float* __restrict__ X, const float* __restrict__ TLE,
    const float* __restrict__ W0, const float* __restrict__ W1,
    const float* __restrict__ W2, const float* __restrict__ W3,
    const float* __restrict__ W4, const float* __restrict__ W5,
    const float* __restrict__ B0, const float* __restrict__ B1,
    const float* __restrict__ B2, const float* __restrict__ B3,
    const float* __restrict__ B4, const float* __restrict__ B5,
    float* __restrict__ O0, float* __restrict__ O1, float* __restrict__ O2,
    float* __restrict__ O3, float* __restrict__ O4, float* __restrict__ O5) {
  __shared__ float xe[16][KIN];  // 12 KB

  const int tile = blockIdx.x;
  const int tid = threadIdx.x;

  // async stage of the 16x192 XE tile (16 floats per thread, 4x B128)
  {
    int row = tid / 12;
    int seg = tid % 12;                // 12 segments of 16 floats
    size_t m = (size_t)tile * 16 + row;
    const float* src = (seg < 4) ? (X + m * DD + seg * 16)
                                 : (TLE + m * (2 * DD) + (seg - 4) * 16);
    float* dst = &xe[row][seg * 16];
#pragma unroll
    for (int q = 0; q < 4; ++q) async_b128(dst + 4 * q, src + 4 * q);
  }
  async_wait0();
  __syncthreads();

  const int w = tid >> 5;  // wave id == projection id
  const int l = tid & 31;
  const int m = l & 15;
  const int kh = (l < 16) ? 0 : 2;

  const float* W = (w == 0) ? W0 : (w == 1) ? W1 : (w == 2) ? W2
                 : (w == 3) ? W3 : (w == 4) ? W4 : W5;
  const float* Bv = (w == 0) ? B0 : (w == 1) ? B1 : (w == 2) ? B2
                  : (w == 3) ? B3 : (w == 4) ? B4 : B5;
  float* O = (w == 0) ? O0 : (w == 1) ? O1 : (w == 2) ? O2
           : (w == 3) ? O3 : (w == 4) ? O4 : O5;

  v8f acc[4] = {};
  for (int s = 0; s < KIN / 4; ++s) {
    int k0 = 4 * s + kh;
    v2f a; a.x = xe[m][k0]; a.y = xe[m][k0 + 1];
#pragma unroll
    for (int nt = 0; nt < 4; ++nt) {
      int n = nt * 16 + m;
      v2f b; b.x = W[(size_t)k0 * DD + n]; b.y = W[(size_t)(k0 + 1) * DD + n];
      acc[nt] = wmma4(a, b, acc[nt]);
    }
  }
  size_t rb = (size_t)tile * 16;
#pragma unroll
  for (int nt = 0; nt < 4; ++nt) {
    int n = nt * 16 + m;
    float bn = Bv[n];
#pragma unroll
    for (int r = 0; r < 8; ++r) {
      int row = r + ((l < 16) ? 0 : 8);
      float v = acc[nt][r] + bn;
      O[(rb + row) * DD + n] = v > 0.f ? v : 0.f;
    }
  }
}

// =====================================================================
// Kernel 2: spatial attention over N=128 nodes. grid = B*T = 384 blocks,
// block = 256 threads (8 waves, one head each).
// =====================================================================
__global__ __launch_bounds__(256) void spatial_kernel(
    const float* __restrict__ Q, const float* __restrict__ K,
    const float* __restrict__ V, float* __restrict__ HS) {
  __shared__ float strip[8][16][NN];  // 64 KB: one 16x128 score strip per head

  const int bt = blockIdx.x;
  const int h = threadIdx.x >> 5;
  const int l = threadIdx.x & 31;
  const int m = l & 15;
  const int kh = (l < 16) ? 0 : 2;
  const size_t base = (size_t)bt * NN;  // first row of this (b,t)
  float(*S)[NN] = strip[h];

  for (int mt = 0; mt < 8; ++mt) {
    // ---- scores: Q(16x8) x K^T(8x128) ----
    v8f acc[8] = {};
#pragma unroll
    for (int s = 0; s < 2; ++s) {
      int k0 = 4 * s + kh;
      v2f a = *(const v2f*)&Q[(base + mt * 16 + m) * DD + h * 8 + k0];
#pragma unroll
      for (int nt = 0; nt < 8; ++nt) {
        v2f b = *(const v2f*)&K[(base + nt * 16 + m) * DD + h * 8 + k0];
        acc[nt] = wmma4(a, b, acc[nt]);
      }
    }
#pragma unroll
    for (int nt = 0; nt < 8; ++nt)
#pragma unroll
      for (int r = 0; r < 8; ++r) {
        int row = r + ((l < 16) ? 0 : 8);
        S[row][nt * 16 + m] = acc[nt][r] * SCALE;
      }
    __syncthreads();

    // ---- row softmax (each lane owns row l%16; halves duplicate) ----
    {
      float* rowp = S[m];
      float mx = -3.4e38f;
      for (int j = 0; j < NN; ++j) mx = fmaxf(mx, rowp[j]);
      float sum = 0.f;
      for (int j = 0; j < NN; ++j) { float e = __expf(rowp[j] - mx); rowp[j] = e; sum += e; }
      float inv = 1.f / sum;
      for (int j = 0; j < NN; ++j) rowp[j] *= inv;
    }
    __syncthreads();

    // ---- P(16x128) x V(128x8) : cols 0..7 of the 16x16 C are valid ----
    v8f o = {};
    const int n = m;
    for (int s = 0; s < NN / 4; ++s) {
      int k0 = 4 * s + kh;
      v2f a; a.x = S[m][k0]; a.y = S[m][k0 + 1];
      v2f b;
      b.x = (n < 8) ? V[(base + k0) * DD + h * 8 + n] : 0.f;
      b.y = (n < 8) ? V[(base + k0 + 1) * DD + h * 8 + n] : 0.f;
      o = wmma4(a, b, o);
    }
    if (n < 8) {
#pragma unroll
      for (int r = 0; r < 8; ++r) {
        int row = r + ((l < 16) ? 0 : 8);
        HS[(base + mt * 16 + row) * DD + h * 8 + n] = o[r];
      }
    }
    __syncthreads();
  }
}

// =====================================================================
// Kernel 3: temporal attention over T=24 (causal + key padding).
// grid = B*N = 2048 blocks, block = 256 threads (8 waves = heads).
// =====================================================================
__global__ __launch_bounds__(256) void temporal_kernel(
    const float* __restrict__ Q, const float* __restrict__ K,
    const float* __restrict__ V, const int* __restrict__ kpm,
    float* __restrict__ HT) {
  __shared__ float sbuf[8][32][32];  // 32 KB

  const int bn = blockIdx.x;
  const int b = bn >> 7;     // / NN
  const int nn = bn & 127;   // % NN
  const int h = threadIdx.x >> 5;
  const int l = threadIdx.x & 31;
  const int m = l & 15;
  const int kh = (l < 16) ? 0 : 2;
  const int lim = kpm[b];  // valid keys j < lim  (lim >= 1)
  float(*S)[32] = sbuf[h];
  const size_t rs = (size_t)NN * DD;                    // stride between times
  const size_t qb = ((size_t)b * TT * NN + nn) * DD + h * 8;

  // ---- scores 24x24 via 2x2 WMMA tiles ----
  v8f acc[2][2] = {};
#pragma unroll
  for (int s = 0; s < 2; ++s) {
    int k0 = 4 * s + kh;
    v2f a0 = *(const v2f*)&Q[qb + (size_t)m * rs + k0];
    v2f a1; a1.x = 0.f; a1.y = 0.f;
    if (16 + m < TT) a1 = *(const v2f*)&Q[qb + (size_t)(16 + m) * rs + k0];
#pragma unroll
    for (int nt = 0; nt < 2; ++nt) {
      int c = nt * 16 + m;
      v2f bv; bv.x = 0.f; bv.y = 0.f;
      if (c < TT) bv = *(const v2f*)&K[qb + (size_t)c * rs + k0];
      acc[0][nt] = wmma4(a0, bv, acc[0][nt]);
      acc[1][nt] = wmma4(a1, bv, acc[1][nt]);
    }
  }
#pragma unroll
  for (int mt = 0; mt < 2; ++mt)
#pragma unroll
    for (int nt = 0; nt < 2; ++nt)
#pragma unroll
      for (int r = 0; r < 8; ++r) {
        int row = mt * 16 + r + ((l < 16) ? 0 : 8);
        int col = nt * 16 + m;
        if (row < TT) S[row][col] = acc[mt][nt][r] * SCALE;
      }
  __syncthreads();

  // ---- masked softmax: rows m and m+16 per lane (halves duplicate) ----
  for (int rr = 0; rr < 2; ++rr) {
    int i = m + rr * 16;
    if (i < TT) {
      int jmax = (i < lim - 1) ? i : (lim - 1);  // inclusive upper bound
      float mx = -3.4e38f;
      for (int j = 0; j <= jmax; ++j) mx = fmaxf(mx, S[i][j]);
      float sum = 0.f;
      for (int j = 0; j <= jmax; ++j) { float e = __expf(S[i][j] - mx); S[i][j] = e; sum += e; }
      float inv = 1.f / sum;
      for (int j = 0; j < TT; ++j) S[i][j] = (j <= jmax) ? S[i][j] * inv : 0.f;
    }
  }
  __syncthreads();

  // ---- P(24x24) x V(24x8) ----
  const int n = m;
#pragma unroll
  for (int mt = 0; mt < 2; ++mt) {
    v8f o = {};
    for (int s = 0; s < TT / 4; ++s) {
      int k0 = 4 * s + kh;
      int row = mt * 16 + m;
      v2f a; a.x = 0.f; a.y = 0.f;
      if (row < TT) { a.x = S[row][k0]; a.y = S[row][k0 + 1]; }
      v2f bv;
      bv.x = (n < 8) ? V[qb + (size_t)k0 * rs + n] : 0.f;
      bv.y = (n < 8) ? V[qb + (size_t)(k0 + 1) * rs + n] : 0.f;
      o = wmma4(a, bv, o);
    }
#pragma unroll
    for (int r = 0; r < 8; ++r) {
      int t = mt * 16 + r + ((l < 16) ? 0 : 8);
      if (t < TT && n < 8) HT[qb + (size_t)t * rs + n] = o[r];
    }
  }
}

// =====================================================================
// Kernel 4: output heads + gated fusion + residual.
// grid = PTOT/16 = 3072 blocks, block = 128 threads (4 waves = 4 col-tiles)
// =====================================================================
__global__ __launch_bounds__(128) void fuse_kernel(
    const float* __restrict__ HSp, const float* __restrict__ HTp,
    const float* __restrict__ X,
    const float* __restrict__ saWo1, const float* __restrict__ sabo1,
    const float* __restrict__ saWo2, const float* __restrict__ sabo2,
    const float* __restrict__ taWo1, const float* __restrict__ tabo1,
    const float* __restrict__ taWo2, const float* __restrict__ tabo2,
    const float* __restrict__ Wxs, const float* __restrict__ Wxt,
    const float* __restrict__ bxt, const float* __restrict__ Wh1,
    const float* __restrict__ bh1, const float* __restrict__ Wh2,
    const float* __restrict__ bh2, float* __restrict__ out) {
  __shared__ float bIn[16][DD], bA[16][DD], bBuf[16][DD], bT[16][DD];  // 16 KB

  const int tile = blockIdx.x;
  const int tid = threadIdx.x;
  const int w = tid >> 5;  // col-tile
  const int l = tid & 31;
  const int m = l & 15;
  const int kh = (l < 16) ? 0 : 2;
  const size_t rb = (size_t)tile * 16;

  auto loadTile = [&](const float* src, float(*dst)[DD]) {
    // 1024 floats, 128 threads -> 8 floats each via 2 async B128
    int r = tid >> 3;
    int c = (tid & 7) * 8;
    const float* s = &src[(rb + r) * DD + c];
    async_b128(&dst[r][c], s);
    async_b128(&dst[r][c + 4], s + 4);
    async_wait0();
  };
  auto mm = [&](const float(*in)[DD], const float* W, v8f& acc) {
    for (int s = 0; s < DD / 4; ++s) {
      int k0 = 4 * s + kh;
      v2f a; a.x = in[m][k0]; a.y = in[m][k0 + 1];
      int n = w * 16 + m;
      v2f b; b.x = W[(size_t)k0 * DD + n]; b.y = W[(size_t)(k0 + 1) * DD + n];
      acc = wmma4(a, b, acc);
    }
  };
  auto store = [&](float(*dst)[DD], v8f acc, const float* bias, bool relu) {
    int n = w * 16 + m;
    float bn = bias[n];
#pragma unroll
    for (int r = 0; r < 8; ++r) {
      int row = r + ((l < 16) ? 0 : 8);
      float v = acc[r] + bn;
      dst[row][n] = relu ? fmaxf(v, 0.f) : v;
    }
  };

  loadTile(HSp, bIn); __syncthreads();
  { v8f a = {}; mm(bIn, saWo1, a); store(bT, a, sabo1, true); } __syncthreads();
  { v8f a = {}; mm(bT, saWo2, a); store(bA, a, sabo2, false); } __syncthreads();
  loadTile(HTp, bIn); __syncthreads();
  { v8f a = {}; mm(bIn, taWo1, a); store(bT, a, tabo1, true); } __syncthreads();
  { v8f a = {}; mm(bT, taWo2, a); store(bBuf, a, tabo2, false); } __syncthreads();
  // gated fusion: z = sigmoid(HS@Wxs + HT@Wxt + bxt); H = z*HS + (1-z)*HT
  {
    v8f a = {};
    mm(bA, Wxs, a);
    mm(bBuf, Wxt, a);
    int n = w * 16 + m;
    float bn = bxt[n];
#pragma unroll
    for (int r = 0; r < 8; ++r) {
      int row = r + ((l < 16) ? 0 : 8);
      float z = 1.f / (1.f + __expf(-(a[r] + bn)));
      bT[row][n] = z * bA[row][n] + (1.f - z) * bBuf[row][n];
    }
  }
  __syncthreads();
  { v8f a = {}; mm(bT, Wh1, a); store(bIn, a, bh1, true); } __syncthreads();
  {
    v8f a = {};
    mm(bIn, Wh2, a);
    int n = w * 16 + m;
    float bn = bh2[n];
#pragma unroll
    for (int r = 0; r < 8; ++r) {
      int row = r + ((l < 16) ? 0 : 8);
      out[(rb + row) * DD + n] = X[(rb + row) * DD + n] + a[r] + bn;
    }
  }
}

// =====================================================================
extern "C" void kernel_launch(void* const* d_in, const int* in_sizes, int n_in,
                              void* d_out, int out_size, void* d_ws,
                              size_t ws_size, hipStream_t stream) {
  const float* X   = (const float*)d_in[0];
  const float* TLE = (const float*)d_in[1];
  const int* kpm   = (const int*)d_in[2];
  const float* sa_Wq  = (const float*)d_in[3];
  const float* sa_bq  = (const float*)d_in[4];
  const float* sa_Wk  = (const float*)d_in[5];
  const float* sa_bk  = (const float*)d_in[6];
  const float* sa_Wv  = (const float*)d_in[7];
  const float* sa_bv  = (const float*)d_in[8];
  const float* sa_Wo1 = (const float*)d_in[9];
  const float* sa_bo1 = (const float*)d_in[10];
  const float* sa_Wo2 = (const float*)d_in[11];
  const float* sa_bo2 = (const float*)d_in[12];
  const float* ta_Wq  = (const float*)d_in[13];
  const float* ta_bq  = (const float*)d_in[14];
  const float* ta_Wk  = (const float*)d_in[15];
  const float* ta_bk  = (const float*)d_in[16];
  const float* ta_Wv  = (const float*)d_in[17];
  const float* ta_bv  = (const float*)d_in[18];
  const float* ta_Wo1 = (const float*)d_in[19];
  const float* ta_bo1 = (const float*)d_in[20];
  const float* ta_Wo2 = (const float*)d_in[21];
  const float* ta_bo2 = (const float*)d_in[22];
  const float* g_Wxs  = (const float*)d_in[23];
  const float* g_Wxt  = (const float*)d_in[24];
  const float* g_bxt  = (const float*)d_in[25];
  const float* g_Wh1  = (const float*)d_in[26];
  const float* g_bh1  = (const float*)d_in[27];
  const float* g_Wh2  = (const float*)d_in[28];
  const float* g_bh2  = (const float*)d_in[29];

  // workspace: 8 regions of P*64 fp32 (~100.6 MB total)
  float* ws = (float*)d_ws;
  const size_t PD = (size_t)PTOT * DD;
  float* qS = ws + 0 * PD; float* kS = ws + 1 * PD; float* vS = ws + 2 * PD;
  float* qT = ws + 3 * PD; float* kT = ws + 4 * PD; float* vT = ws + 5 * PD;
  float* hsp = ws + 6 * PD; float* htp = ws + 7 * PD;

  proj_kernel<<<PTOT / 16, 192, 0, stream>>>(
      X, TLE, sa_Wq, sa_Wk, sa_Wv, ta_Wq, ta_Wk, ta_Wv, sa_bq, sa_bk, sa_bv,
      ta_bq, ta_bk, ta_bv, qS, kS, vS, qT, kT, vT);
  spatial_kernel<<<BB * TT, 256, 0, stream>>>(qS, kS, vS, hsp);
  temporal_kernel<<<BB * NN, 256, 0, stream>>>(qT, kT, vT, kpm, htp);
  fuse_kernel<<<PTOT / 16, 128, 0, stream>>>(
      hsp, htp, X, sa_Wo1, sa_bo1, sa_Wo2, sa_bo2, ta_Wo1, ta_bo1, ta_Wo2,
      ta_bo2, g_Wxs, g_Wxt, g_bxt, g_Wh1, g_bh1, g_Wh2, g_bh2, (float*)d_out);
}